// PointNetFeaturePropagation_41540923687665
// MI455X (gfx1250) — compile-verified
//
#include <hip/hip_runtime.h>

typedef float v2f __attribute__((ext_vector_type(2)));
typedef float v8f __attribute__((ext_vector_type(8)));

#define EPS_DIST 1e-8f
#define EPS_BN   1e-5f

#define B_   16
#define N_   4096
#define M_   1024
#define C1_  128
#define C2_  256
#define K0_  (C1_ + C2_)   // 384
#define O0_  256
#define K1_  256
#define O1_  128

// GEMM tiling
#define BN_TILE 64          // n columns per block
#define BO_TILE 128         // o rows per block (8 waves x 16)
#define BK_TILE 32          // k chunk per staging round
// X panel: K-pair interleaved layout. Element (k,n) -> Xs[(k>>1)*XP_STRIDE + 2n + (k&1)].
// XP_STRIDE = 160 = 2*BN_TILE + 32 pad; 160 mod 64 == 32 -> the two lane-halves
// (kp even / odd) read disjoint 32-bank sets: conflict-free ds_load_b64.
#define XP_STRIDE 160
#define WS_STRIDE 36        // 36r mod 64 distinct for r=0..15; keeps 16B alignment for b128 stores

// ---------------------------------------------------------------------------
// Kernel 1: 3-NN search. One lane per query point n; xyz2[b] staged in LDS.
// Top-3 tracked on squared distance (monotonic under sqrt).
// ---------------------------------------------------------------------------
__global__ void knn3_kernel(const float* __restrict__ xyz1,
                            const float* __restrict__ xyz2,
                            int* __restrict__ idxb,
                            float* __restrict__ wb) {
    __shared__ float sx[M_ * 3];
    const int b = blockIdx.y;
    for (int i = threadIdx.x; i < M_ * 3; i += blockDim.x)
        sx[i] = xyz2[(size_t)b * M_ * 3 + i];
    __syncthreads();

    const int n = blockIdx.x * blockDim.x + threadIdx.x;
    const size_t p = ((size_t)b * N_ + n) * 3;
    const float px = xyz1[p + 0];
    const float py = xyz1[p + 1];
    const float pz = xyz1[p + 2];

    float bd0 = 3.4e38f, bd1 = 3.4e38f, bd2 = 3.4e38f;
    int   bi0 = 0, bi1 = 0, bi2 = 0;
    for (int m = 0; m < M_; ++m) {
        const float dx = px - sx[m * 3 + 0];
        const float dy = py - sx[m * 3 + 1];
        const float dz = pz - sx[m * 3 + 2];
        const float d2 = dx * dx + dy * dy + dz * dz;
        if (d2 < bd2) {
            if (d2 < bd0)      { bd2 = bd1; bi2 = bi1; bd1 = bd0; bi1 = bi0; bd0 = d2; bi0 = m; }
            else if (d2 < bd1) { bd2 = bd1; bi2 = bi1; bd1 = d2;  bi1 = m; }
            else               { bd2 = d2;  bi2 = m; }
        }
    }
    const float w0 = 1.f / (sqrtf(fmaxf(bd0, 0.f)) + EPS_DIST);
    const float w1 = 1.f / (sqrtf(fmaxf(bd1, 0.f)) + EPS_DIST);
    const float w2 = 1.f / (sqrtf(fmaxf(bd2, 0.f)) + EPS_DIST);
    const float inv = 1.f / (w0 + w1 + w2);

    const size_t o = ((size_t)b * N_ + n) * 3;
    idxb[o + 0] = bi0; idxb[o + 1] = bi1; idxb[o + 2] = bi2;
    wb[o + 0] = w0 * inv; wb[o + 1] = w1 * inv; wb[o + 2] = w2 * inv;
}

// ---------------------------------------------------------------------------
// Kernel 2: weighted gather-interpolation. Coalesced over n; random m-gathers
// land in L2 (points2 is 16 MB << 192 MB L2).
// ---------------------------------------------------------------------------
__global__ void interp3_kernel(const float* __restrict__ points2,
                               const int* __restrict__ idxb,
                               const float* __restrict__ wb,
                               float* __restrict__ interp) {
    const int n = blockIdx.x * blockDim.x + threadIdx.x;
    const int c = blockIdx.y;
    const int b = blockIdx.z;
    const size_t o = ((size_t)b * N_ + n) * 3;
    const int i0 = idxb[o + 0], i1 = idxb[o + 1], i2 = idxb[o + 2];
    const float w0 = wb[o + 0], w1 = wb[o + 1], w2 = wb[o + 2];
    const float* row = points2 + ((size_t)b * C2_ + c) * M_;
    const float v = w0 * row[i0] + w1 * row[i1] + w2 * row[i2];
    interp[((size_t)b * C2_ + c) * N_ + n] = v;
}

// ---------------------------------------------------------------------------
// Kernels 3/4: LDS-tiled fp32 WMMA GEMM + fused bias + BN(eval) + ReLU.
//
// Block (256 thr = 8 waves) computes a 128(o) x 64(n) tile. Per 32-wide K
// chunk: X panel (32 x 64, K-pair interleaved) and W panel (128 x 32) staged
// to LDS; the concat split (points1 | interp) is resolved per chunk by a
// UNIFORM scalar branch (ksplit is a multiple of BK_TILE). Each wave owns one
// 16-o subtile and 4 accumulators; per K-step: 1 ds_b64 (A) + 4 ds_b64 (B)
// + 4 WMMA, with B operands landing directly in aligned VGPR pairs.
//
// WMMA f32 16x16x4 VGPR layouts (ISA 7.12.2):
//   A(16x4): lane%16 = M(=o),  K = vgpr + 2*(lane>=16)  -> K pair contiguous in W row
//   B(4x16): lane%16 = N(=n),  K = vgpr + 2*(lane>=16)  -> K pair contiguous in Xs layout
//   C/D:     row M = vgpr + 8*(lane>=16), col N = lane%16
// ---------------------------------------------------------------------------
__global__ __launch_bounds__(256)
void wmma_gemm_bn_relu_kernel(const float* __restrict__ Xa,   // k <  ksplit, (B, ksplit,   N)
                              const float* __restrict__ Xb,   // k >= ksplit, (B, K-ksplit, N)
                              const float* __restrict__ W,    // (O, K)
                              const float* __restrict__ bias,
                              const float* __restrict__ gamma,
                              const float* __restrict__ beta,
                              const float* __restrict__ mean,
                              const float* __restrict__ var,
                              float* __restrict__ Y,          // (B, O, N)
                              int K, int ksplit, int O) {
    __shared__ float Xs[(BK_TILE / 2) * XP_STRIDE];
    __shared__ float Ws[BO_TILE * WS_STRIDE];

    const int tid  = threadIdx.x;
    const int wave = tid >> 5;        // 0..7 -> o subtile
    const int lane = tid & 31;
    const int half = lane >> 4;
    const int lq   = lane & 15;
    const int nb0  = blockIdx.x * BN_TILE;
    const int ob0  = blockIdx.y * BO_TILE;
    const int b    = blockIdx.z;

    // staging maps
    const int xr = tid >> 4;              // 0..15 (X k-row; +16 on 2nd pass)
    const int xc = (tid & 15) * 4;        // 0..60
    const int wr = tid >> 3;              // 0..31 (W row; +32 per pass)
    const int wc = (tid & 7) * 4;         // 0..28

    v8f acc0 = {0,0,0,0,0,0,0,0};
    v8f acc1 = {0,0,0,0,0,0,0,0};
    v8f acc2 = {0,0,0,0,0,0,0,0};
    v8f acc3 = {0,0,0,0,0,0,0,0};

    for (int kk0 = 0; kk0 < K; kk0 += BK_TILE) {
        // ---- choose source tensor for this K chunk (uniform scalar branch) ----
        const float* xbase = (kk0 < ksplit)
            ? (Xa + ((size_t)b * ksplit + kk0) * N_)
            : (Xb + ((size_t)b * (K - ksplit) + (kk0 - ksplit)) * N_);

        // ---- stage X panel: BK_TILE x BN_TILE, K-pair interleaved ----
#pragma unroll
        for (int rr = 0; rr < BK_TILE; rr += 16) {
            const int k = xr + rr;
            const float4 v = *(const float4*)(xbase + (size_t)k * N_ + nb0 + xc);
            float* d = &Xs[(k >> 1) * XP_STRIDE + 2 * xc + (k & 1)];
            d[0] = v.x; d[2] = v.y; d[4] = v.z; d[6] = v.w;
        }
        // ---- stage W panel: BO_TILE x BK_TILE ----
#pragma unroll
        for (int rr = 0; rr < BO_TILE; rr += 32) {
            const float4 v = *(const float4*)(W + (size_t)(ob0 + wr + rr) * K + kk0 + wc);
            float* d = &Ws[(wr + rr) * WS_STRIDE + wc];
            d[0] = v.x; d[1] = v.y; d[2] = v.z; d[3] = v.w;
        }
        // prefetch next X chunk toward L2 while we compute (global_prefetch_b8)
        if (kk0 + BK_TILE < K) {
            const float* nxt = (kk0 + BK_TILE < ksplit)
                ? (Xa + ((size_t)b * ksplit + kk0 + BK_TILE) * N_)
                : (Xb + ((size_t)b * (K - ksplit) + (kk0 + BK_TILE - ksplit)) * N_);
            __builtin_prefetch(nxt + (size_t)xr * N_ + nb0 + xc, 0, 1);
        }
        __syncthreads();

        // ---- compute: 8 K-steps x 4 WMMA ----
#pragma unroll
        for (int kk = 0; kk < BK_TILE; kk += 4) {
            const int ka = kk + 2 * half;               // K held in vgpr0 of this lane
            const int kp = (kk >> 1) + half;            // K-pair row in Xs
            const v2f av = *(const v2f*)&Ws[(wave * 16 + lq) * WS_STRIDE + ka];

            const float* xrow = &Xs[kp * XP_STRIDE + 2 * lq];
            const v2f bv0 = *(const v2f*)(xrow +  0);
            const v2f bv1 = *(const v2f*)(xrow + 32);
            const v2f bv2 = *(const v2f*)(xrow + 64);
            const v2f bv3 = *(const v2f*)(xrow + 96);

            acc0 = __builtin_amdgcn_wmma_f32_16x16x4_f32(false, av, false, bv0, (short)0, acc0, false, false);
            acc1 = __builtin_amdgcn_wmma_f32_16x16x4_f32(false, av, false, bv1, (short)0, acc1, false, false);
            acc2 = __builtin_amdgcn_wmma_f32_16x16x4_f32(false, av, false, bv2, (short)0, acc2, false, false);
            acc3 = __builtin_amdgcn_wmma_f32_16x16x4_f32(false, av, false, bv3, (short)0, acc3, false, false);
        }
        __syncthreads();
    }

    // ---- epilogue: y = acc * s + shift, ReLU, store ----
    const int orow = ob0 + wave * 16;
#pragma unroll
    for (int v = 0; v < 8; ++v) {
        const int o = orow + v + 8 * half;
        const float s  = gamma[o] * rsqrtf(var[o] + EPS_BN);
        const float sh = (bias[o] - mean[o]) * s + beta[o];
        float* yrow = Y + ((size_t)b * O + o) * N_ + nb0 + lq;
        yrow[ 0] = fmaxf(acc0[v] * s + sh, 0.f);
        yrow[16] = fmaxf(acc1[v] * s + sh, 0.f);
        yrow[32] = fmaxf(acc2[v] * s + sh, 0.f);
        yrow[48] = fmaxf(acc3[v] * s + sh, 0.f);
    }
}

// ---------------------------------------------------------------------------
// Launch
// ---------------------------------------------------------------------------
extern "C" void kernel_launch(void* const* d_in, const int* in_sizes, int n_in,
                              void* d_out, int out_size, void* d_ws, size_t ws_size,
                              hipStream_t stream) {
    const float* xyz1    = (const float*)d_in[0];
    const float* xyz2    = (const float*)d_in[1];
    const float* points1 = (const float*)d_in[2];
    const float* points2 = (const float*)d_in[3];
    const float* w0      = (const float*)d_in[4];
    const float* b0      = (const float*)d_in[5];
    const float* g0      = (const float*)d_in[6];
    const float* beta0   = (const float*)d_in[7];
    const float* m0      = (const float*)d_in[8];
    const float* v0      = (const float*)d_in[9];
    const float* w1      = (const float*)d_in[10];
    const float* b1      = (const float*)d_in[11];
    const float* g1      = (const float*)d_in[12];
    const float* beta1   = (const float*)d_in[13];
    const float* m1      = (const float*)d_in[14];
    const float* v1      = (const float*)d_in[15];

    char* ws = (char*)d_ws;
    size_t off = 0;
    int*   idxb   = (int*)(ws + off);   off += (size_t)B_ * N_ * 3 * sizeof(int);
    float* wb     = (float*)(ws + off); off += (size_t)B_ * N_ * 3 * sizeof(float);
    off = (off + 255) & ~(size_t)255;
    float* interp = (float*)(ws + off); off += (size_t)B_ * C2_ * N_ * sizeof(float);
    float* y0     = (float*)(ws + off); off += (size_t)B_ * O0_ * N_ * sizeof(float);

    // 1) 3-NN search
    knn3_kernel<<<dim3(N_ / 256, B_), 256, 0, stream>>>(xyz1, xyz2, idxb, wb);

    // 2) inverse-distance interpolation of points2 -> (B, C2, N)
    interp3_kernel<<<dim3(N_ / 256, C2_, B_), 256, 0, stream>>>(points2, idxb, wb, interp);

    // 3) conv0 (384->256) + BN + ReLU ; K split across points1 | interp
    wmma_gemm_bn_relu_kernel<<<dim3(N_ / BN_TILE, O0_ / BO_TILE, B_), 256, 0, stream>>>(
        points1, interp, w0, b0, g0, beta0, m0, v0, y0, K0_, C1_, O0_);

    // 4) conv1 (256->128) + BN + ReLU -> d_out
    wmma_gemm_bn_relu_kernel<<<dim3(N_ / BN_TILE, O1_ / BO_TILE, B_), 256, 0, stream>>>(
        y0, y0, w1, b1, g1, beta1, m1, v1, (float*)d_out, K1_, K1_, O1_);
}